// QBGating_41205916238370
// MI455X (gfx1250) — compile-verified
//
#include <hip/hip_runtime.h>
#include <hip/hip_bf16.h>

// Quantile-balanced sparse gating forward for MI455X (gfx1250, wave32).
// One row per wave32; 8 waves (256 threads) per block.
// - Async global->LDS copy engine (ASYNCcnt path) streams BOTH the logits tile
//   and the beta vector into LDS; one s_wait_asynccnt + one barrier.
// - Iterative top-8 via per-lane argmax + __shfl_xor butterfly (wave32).
// - Non-temporal 128-bit zero stores + NT scalar scatter (per-wave store order).

#define K_EFF     8
#define MCOLS     512
#define WAVES     8
#define BLOCK_T   (WAVES * 32)

typedef float v4f __attribute__((ext_vector_type(4)));

#if defined(__AMDGCN__) && \
    __has_builtin(__builtin_amdgcn_global_load_async_to_lds_b128) && \
    __has_builtin(__builtin_amdgcn_s_wait_asynccnt)
#define USE_ASYNC_LDS 1
typedef int v4i __attribute__((vector_size(16)));
typedef __attribute__((address_space(1))) v4i* g_v4i_p;   // global (AS1)
typedef __attribute__((address_space(3))) v4i* l_v4i_p;   // LDS (AS3)
#else
#define USE_ASYNC_LDS 0
#endif

__global__ __launch_bounds__(BLOCK_T)
void QBGating_topk_kernel(const float* __restrict__ logits,
                          const float* __restrict__ beta,
                          float* __restrict__ out,
                          int nrows)
{
    const int lane = threadIdx.x & 31;
    const int wave = threadIdx.x >> 5;
    const int row  = blockIdx.x * WAVES + wave;
    const bool row_ok = (row < nrows);

    __shared__ __align__(16) float s_beta[MCOLS];
#if USE_ASYNC_LDS
    __shared__ __align__(16) float s_tile[WAVES * MCOLS];
#endif

    const float* __restrict__ grow = logits + (size_t)row * MCOLS;

#if USE_ASYNC_LDS
    // Stage beta (2KB): threads 0..127 (waves 0..3, wave-uniform) issue one
    // 16B async chunk each through the copy engine.
    if (threadIdx.x < (MCOLS / 4)) {
        __builtin_amdgcn_global_load_async_to_lds_b128(
            (g_v4i_p)(void*)(beta + threadIdx.x * 4),
            (l_v4i_p)(void*)(s_beta + threadIdx.x * 4),
            /*offset=*/0, /*cpol=*/0);
    }
    // Stage this wave's logits row (2KB): 4 x 16B per lane.
    if (row_ok) {
        float* lrow = s_tile + wave * MCOLS;
        #pragma unroll
        for (int c = 0; c < 4; ++c) {
            __builtin_amdgcn_global_load_async_to_lds_b128(
                (g_v4i_p)(void*)(grow + c * 128 + lane * 4),
                (l_v4i_p)(void*)(lrow + c * 128 + lane * 4),
                /*offset=*/0, /*cpol=*/0);
        }
    }
    __builtin_amdgcn_s_wait_asynccnt(0);   // this wave's DMA writes landed in LDS
    __syncthreads();                       // publish beta to all waves
#else
    if (threadIdx.x < (MCOLS / 4))
        *(v4f*)(s_beta + threadIdx.x * 4) = *(const v4f*)(beta + threadIdx.x * 4);
    __syncthreads();
#endif

    if (!row_ok) return;

    // Each lane owns 16 columns: col = c*128 + lane*4 + e  (c,e in 0..3).
    // Keep only the debiased score in registers; raw logit is recovered
    // later as adj + beta[col] (uniform LDS broadcast read).
    float adj[16];
    #pragma unroll
    for (int c = 0; c < 4; ++c) {
        v4f rv;
#if USE_ASYNC_LDS
        rv = *(const v4f*)(s_tile + wave * MCOLS + c * 128 + lane * 4);
#else
        rv = __builtin_nontemporal_load((const v4f*)(grow + c * 128 + lane * 4));
#endif
        v4f bv = *(const v4f*)(s_beta + c * 128 + lane * 4);
        #pragma unroll
        for (int e = 0; e < 4; ++e)
            adj[c * 4 + e] = rv[e] - bv[e];
    }

    const float NEG = -__builtin_inff();
    float sel_raw[K_EFF];   // uniform across lanes (constant-indexed only)
    int   my_col = 0;       // lane j (< K_EFF) remembers winner j's column

    #pragma unroll
    for (int j = 0; j < K_EFF; ++j) {
        // ---- local argmax over this lane's 16 elements ----
        // iteration order is strictly increasing column -> lowest-index tie-break
        float bv = NEG;
        int   bc = 0;
        #pragma unroll
        for (int c = 0; c < 4; ++c) {
            #pragma unroll
            for (int e = 0; e < 4; ++e) {
                const int   i    = c * 4 + e;
                const int   colI = (c << 7) | (lane << 2) | e;
                const bool  t    = adj[i] > bv;
                bv = t ? adj[i] : bv;
                bc = t ? colI   : bc;
            }
        }
        // ---- wave32 butterfly reduction; prefer lower column on ties ----
        #pragma unroll
        for (int off = 16; off > 0; off >>= 1) {
            const float ov = __shfl_xor(bv, off, 32);
            const int   oc = __shfl_xor(bc, off, 32);
            const bool take = (ov > bv) || ((ov == bv) && (oc < bc));
            bv = take ? ov : bv;
            bc = take ? oc : bc;
        }
        // (bv, bc) now uniform across the wave.
        sel_raw[j] = bv + s_beta[bc];           // recover raw logit (broadcast)
        my_col = (lane == j) ? bc : my_col;     // lane j will do the scatter

        // ---- mask the winner out of this lane's registers (no dyn indexing) ----
        const int rel = bc - (lane << 2);       // (c<<7)|e if this lane owns it
        #pragma unroll
        for (int c = 0; c < 4; ++c) {
            #pragma unroll
            for (int e = 0; e < 4; ++e) {
                const int i = c * 4 + e;
                adj[i] = (rel == ((c << 7) | e)) ? NEG : adj[i];
            }
        }
    }

    // ---- softmax over the 8 selected raw logits (uniform, redundant) ----
    float m = sel_raw[0];
    #pragma unroll
    for (int j = 1; j < K_EFF; ++j) m = fmaxf(m, sel_raw[j]);
    float sum = 0.0f;
    float p[K_EFF];
    #pragma unroll
    for (int j = 0; j < K_EFF; ++j) { p[j] = expf(sel_raw[j] - m); sum += p[j]; }
    const float inv = 1.0f / sum;
    float my_p = 0.0f;
    #pragma unroll
    for (int j = 0; j < K_EFF; ++j) my_p = (lane == j) ? p[j] * inv : my_p;

    // ---- dense output: NT zero-fill, then NT scatter of 8 probabilities.
    // Per-wave stores retire in order, so the scatter wins over the zeros.
    float* __restrict__ orow = out + (size_t)row * MCOLS;
    const v4f z = {0.0f, 0.0f, 0.0f, 0.0f};
    #pragma unroll
    for (int c = 0; c < 4; ++c)
        __builtin_nontemporal_store(z, (v4f*)(orow + c * 128 + lane * 4));
    if (lane < K_EFF)
        __builtin_nontemporal_store(my_p, orow + my_col);
}

extern "C" void kernel_launch(void* const* d_in, const int* in_sizes, int n_in,
                              void* d_out, int out_size, void* d_ws, size_t ws_size,
                              hipStream_t stream) {
    const float* logits = (const float*)d_in[0];   // [N, 512] fp32
    const float* beta   = (const float*)d_in[1];   // [512]    fp32
    float*       out    = (float*)d_out;           // [N, 512] fp32
    const int N = in_sizes[0] / MCOLS;

    const dim3 grid((N + WAVES - 1) / WAVES);
    const dim3 block(BLOCK_T);
    QBGating_topk_kernel<<<grid, block, 0, stream>>>(logits, beta, out, N);
}